// GCN_50431505990378
// MI455X (gfx1250) — compile-verified
//
#include <hip/hip_runtime.h>

typedef float        v2f __attribute__((ext_vector_type(2)));
typedef float        v8f __attribute__((ext_vector_type(8)));
typedef unsigned int v4u __attribute__((ext_vector_type(4)));
typedef int          v8i __attribute__((ext_vector_type(8)));
typedef int          v4i __attribute__((ext_vector_type(4)));

#define N_NODES 50000
#define N_EDGES 300000

// ---------------------------------------------------------------------------
// GEMM: S[M,N] = H[M,K] * W[K,N]   (fp32, V_WMMA_F32_16X16X4_F32)
// block = 256 threads = 8 waves. Block tile 128(M) x 64(N).
// Each wave computes a 16x64 strip = 4 WMMA accumulators, K-loop step 4.
// W tile (K x 64, row-major, <=64KB) is staged into LDS by the Tensor Data
// Mover (TENSOR_LOAD_TO_LDS, TENSORcnt), issued by wave 0 only.
// B fragment j at k0: lanes<16 need rows k0,k0+1; lanes>=16 rows k0+2,k0+3.
// The two halves are 64 dwords apart -> one DS_LOAD_2ADDR_B32 per fragment,
// emitted via inline asm so the pairing lands directly in the WMMA operand
// register pair (the compiler's DS combiner would otherwise pair by sorted
// offsets and insert v_mov shuffles).
// ---------------------------------------------------------------------------
__global__ __launch_bounds__(256) void gcn_gemm_wmma(
    const float* __restrict__ H, const float* __restrict__ W,
    float* __restrict__ S, int M, int K, int N) {
  __shared__ float ldsB[256 * 64];  // max K=256 rows x 64 cols, row-major

  const int tid  = threadIdx.x;
  const int wid  = tid >> 5;
  const int lane = tid & 31;
  const int half = lane >> 4;   // 0: lanes 0-15, 1: lanes 16-31
  const int l16  = lane & 15;

  const int n0 = blockIdx.y * 64;
  int m0 = blockIdx.x * 128 + wid * 16;
  if (m0 + 16 > M) m0 = M - 16;  // clamp: duplicate work, identical values

  const unsigned ldsBase = (unsigned)(uintptr_t)&ldsB[0];  // low 32 bits = LDS offset

  if (wid == 0) {
    // Tensor DMA descriptor (D#) for a 2D tile: K rows x 64 cols of W[:,n0..]
    const unsigned long long ga = (unsigned long long)(uintptr_t)(W + n0);
    v4u g0;
    g0[0] = 1u;                                   // count=1 user descriptor
    g0[1] = ldsBase;                              // lds_addr
    g0[2] = (unsigned)(ga & 0xFFFFFFFFu);         // global_addr[31:0]
    g0[3] = (unsigned)((ga >> 32) & 0x01FFFFFFu)  // global_addr[56:32]
            | (2u << 30);                         // type = 2 (image)
    v8i g1;
    g1[0] = (2 << 16);                            // wg_mask=0, data_size=2 (4B)
    g1[1] = (int)(((unsigned)N & 0xFFFFu) << 16); // tensor_dim0[15:0]
    g1[2] = (int)((((unsigned)N >> 16) & 0xFFFFu) // tensor_dim0[31:16]
            | (((unsigned)K & 0xFFFFu) << 16));   // tensor_dim1[15:0]
    g1[3] = (int)((((unsigned)K >> 16) & 0xFFFFu) // tensor_dim1[31:16]
            | (64u << 16));                       // tile_dim0 = 64
    g1[4] = K;                                    // tile_dim1=K, tile_dim2=0
    g1[5] = N;                                    // tensor_dim0_stride[31:0]
    g1[6] = 0;                                    // stride hi / dim1_stride lo
    g1[7] = 0;
    __builtin_amdgcn_tensor_load_to_lds(g0, g1, (v4i)0, (v4i)0, (v8i)0, 0);
    __builtin_amdgcn_s_wait_tensorcnt(0);
  }
  __syncthreads();

  v8f acc0 = {}; v8f acc1 = {}; v8f acc2 = {}; v8f acc3 = {};
  const float* arow = H + (size_t)(m0 + l16) * K;
  // Per-lane LDS byte address of this half-wave's first B element.
  const unsigned baddr0 = ldsBase + (unsigned)(l16 * 4 + half * 2 * 256);

  for (int k0 = 0; k0 < K; k0 += 4) {
    // A fragment 16x4: lanes 0-15 -> k0+{0,1}, lanes 16-31 -> k0+{2,3}
    v2f a = *(const v2f*)(arow + k0 + half * 2);
    const unsigned bk = baddr0 + (unsigned)(k0 * 256);  // 64 floats per k row
    v2f b0, b1, b2, b3;
    asm volatile(
        "ds_load_2addr_b32 %0, %4 offset0:0 offset1:64\n\t"
        "ds_load_2addr_b32 %1, %4 offset0:16 offset1:80\n\t"
        "ds_load_2addr_b32 %2, %4 offset0:32 offset1:96\n\t"
        "ds_load_2addr_b32 %3, %4 offset0:48 offset1:112\n\t"
        "s_wait_dscnt 0x0"
        : "=&v"(b0), "=&v"(b1), "=&v"(b2), "=&v"(b3)
        : "v"(bk));
    acc0 = __builtin_amdgcn_wmma_f32_16x16x4_f32(false, a, false, b0, (short)0, acc0, false, false);
    acc1 = __builtin_amdgcn_wmma_f32_16x16x4_f32(false, a, false, b1, (short)0, acc1, false, false);
    acc2 = __builtin_amdgcn_wmma_f32_16x16x4_f32(false, a, false, b2, (short)0, acc2, false, false);
    acc3 = __builtin_amdgcn_wmma_f32_16x16x4_f32(false, a, false, b3, (short)0, acc3, false, false);
  }

  // C/D layout: VGPR r -> row m0 + r + half*8, lane l16 -> col n0 + j*16 + l16.
  // One 64-bit base computation, then pointer-incremental rows (avoids a
  // v_mul_u64 per row in the epilogue).
  float* p = S + (size_t)(m0 + half * 8) * N + (n0 + l16);
#pragma unroll
  for (int r = 0; r < 8; ++r) {
    p[ 0] = acc0[r];
    p[16] = acc1[r];
    p[32] = acc2[r];
    p[48] = acc3[r];
    p += N;
  }
}

// ---------------------------------------------------------------------------
// Edge scatter: agg[dst] += support[src] * w    (one wave32 per edge)
// float2 row reads; per-float GLOBAL_ATOMIC_ADD_F32 into L2.
// ---------------------------------------------------------------------------
__global__ __launch_bounds__(256) void gcn_scatter(
    const float* __restrict__ S, const int* __restrict__ esrc,
    const int* __restrict__ edst, const float* __restrict__ ew,
    float* __restrict__ agg, int dout) {
  const int gw   = (int)((blockIdx.x * (unsigned)blockDim.x + threadIdx.x) >> 5);
  const int lane = threadIdx.x & 31;
  if (gw >= N_EDGES) return;
  const int   s = esrc[gw];
  const int   d = edst[gw];
  const float w = ew[gw];
  const float* srow = S   + (size_t)s * dout;
  float*       drow = agg + (size_t)d * dout;
  for (int f = lane * 2; f < dout; f += 64) {
    const v2f v = *(const v2f*)(srow + f);
    atomicAdd(&drow[f + 0], v.x * w);
    atomicAdd(&drow[f + 1], v.y * w);
  }
}

// ---------------------------------------------------------------------------
// out = relu(agg + b)   (float4)
// ---------------------------------------------------------------------------
__global__ void gcn_bias_relu(const float4* __restrict__ agg,
                              const float* __restrict__ b,
                              float4* __restrict__ out, int dout) {
  const long long i = blockIdx.x * (long long)blockDim.x + threadIdx.x;
  const long long total = ((long long)N_NODES * dout) >> 2;
  if (i >= total) return;
  const int f = (int)((i * 4) % dout);
  float4 v = agg[i];
  v.x = fmaxf(v.x + b[f + 0], 0.f);
  v.y = fmaxf(v.y + b[f + 1], 0.f);
  v.z = fmaxf(v.z + b[f + 2], 0.f);
  v.w = fmaxf(v.w + b[f + 3], 0.f);
  out[i] = v;
}

__global__ void gcn_add2(const float4* __restrict__ a, const float4* __restrict__ b,
                         float4* __restrict__ o, long long n4) {
  const long long i = blockIdx.x * (long long)blockDim.x + threadIdx.x;
  if (i >= n4) return;
  float4 x = a[i]; const float4 y = b[i];
  x.x += y.x; x.y += y.y; x.z += y.z; x.w += y.w;
  o[i] = x;
}

__global__ void gcn_add3(const float4* __restrict__ a, const float4* __restrict__ b,
                         const float4* __restrict__ c, float4* __restrict__ o, long long n4) {
  const long long i = blockIdx.x * (long long)blockDim.x + threadIdx.x;
  if (i >= n4) return;
  float4 x = a[i]; const float4 y = b[i]; const float4 z = c[i];
  x.x += y.x + z.x; x.y += y.y + z.y; x.z += y.z + z.z; x.w += y.w + z.w;
  o[i] = x;
}

// ---------------------------------------------------------------------------
static void run_layer(const float* h_in, const float* W, const float* b,
                      int din, int dout, float* S, float* AGG, float* out,
                      const int* esrc, const int* edst, const float* ew,
                      hipStream_t stream) {
  dim3 ggrid((N_NODES + 127) / 128, dout / 64);
  gcn_gemm_wmma<<<ggrid, 256, 0, stream>>>(h_in, W, S, N_NODES, din, dout);
  (void)hipMemsetAsync(AGG, 0, (size_t)N_NODES * dout * sizeof(float), stream);
  gcn_scatter<<<(N_EDGES * 32 + 255) / 256, 256, 0, stream>>>(S, esrc, edst, ew, AGG, dout);
  const long long t4 = ((long long)N_NODES * dout) >> 2;
  gcn_bias_relu<<<(int)((t4 + 255) / 256), 256, 0, stream>>>(
      (const float4*)AGG, b, (float4*)out, dout);
}

extern "C" void kernel_launch(void* const* d_in, const int* in_sizes, int n_in,
                              void* d_out, int out_size, void* d_ws, size_t ws_size,
                              hipStream_t stream) {
  const float* x    = (const float*)d_in[0];
  const int*   esrc = (const int*)d_in[1];
  const int*   edst = (const int*)d_in[2];
  const float* ew   = (const float*)d_in[3];
  const float* W[6]; const float* B[6];
  for (int i = 0; i < 6; ++i) { W[i] = (const float*)d_in[4 + 2 * i]; B[i] = (const float*)d_in[5 + 2 * i]; }
  static const int DIN[6]  = {128, 64, 128, 256, 256, 256};
  static const int DOUT[6] = {64, 128, 256, 256, 256, 256};

  const size_t SLAB = (size_t)N_NODES * 256;  // max-width slab (floats)
  float* ws  = (float*)d_ws;
  float* S   = ws;             // GEMM output (support)
  float* AGG = ws + SLAB;      // scatter accumulator
  float* H   = ws + 2 * SLAB;  // rolling layer output
  float* L3  = ws + 3 * SLAB;  // kept for residuals
  float* L4  = ws + 4 * SLAB;
  float* TMP = (float*)d_out;  // residual-sum scratch; final layer overwrites it
  const long long n4 = ((long long)N_NODES * 256) >> 2;
  const int ngrid = (int)((n4 + 255) / 256);

  // l1 = relu(gconv(x,  W1)) -> H
  run_layer(x,  W[0], B[0], DIN[0], DOUT[0], S, AGG, H,  esrc, edst, ew, stream);
  // l2 -> H (l1 dead after its GEMM)
  run_layer(H,  W[1], B[1], DIN[1], DOUT[1], S, AGG, H,  esrc, edst, ew, stream);
  // l3 -> L3
  run_layer(H,  W[2], B[2], DIN[2], DOUT[2], S, AGG, L3, esrc, edst, ew, stream);
  // l4 -> L4
  run_layer(L3, W[3], B[3], DIN[3], DOUT[3], S, AGG, L4, esrc, edst, ew, stream);
  // l5 = relu(gconv(l4 + l3, W5)) -> H
  gcn_add2<<<ngrid, 256, 0, stream>>>((const float4*)L4, (const float4*)L3, (float4*)TMP, n4);
  run_layer(TMP, W[4], B[4], DIN[4], DOUT[4], S, AGG, H,  esrc, edst, ew, stream);
  // l6 = relu(gconv(l5 + l4 + l3, W6)) -> d_out
  gcn_add3<<<ngrid, 256, 0, stream>>>((const float4*)H, (const float4*)L4, (const float4*)L3,
                                      (float4*)TMP, n4);
  run_layer(TMP, W[5], B[5], DIN[5], DOUT[5], S, AGG, (float*)d_out, esrc, edst, ew, stream);
}